// EnvLight_59193239274073
// MI455X (gfx1250) — compile-verified
//
#include <hip/hip_runtime.h>
#include <stdint.h>

// ---------------- constants from the reference ----------------
#define MINR 0.08f
#define MAXR 0.5f
#define NLEV 6
#define DIFF_RES 16
#define DIFF_FLOATS (6 * DIFF_RES * DIFF_RES * 3)   // 4608 floats = 18432 B

struct F3 { float x, y, z; };

// ---------------- cubemap face math (matches reference exactly) -------------
__device__ __forceinline__ void cube_uv(float x, float y, float z, float Rf,
                                        int& face, float& u, float& v) {
  float ax = fabsf(x), ay = fabsf(y), az = fabsf(z);
  bool is_x = (ax >= ay) && (ax >= az);
  bool is_y = (!is_x) && (ay >= az);
  float ma, sc, tc;
  if (is_x)      { face = (x > 0.f) ? 0 : 1; ma = ax; sc = (x > 0.f) ? -z : z; tc = -y; }
  else if (is_y) { face = (y > 0.f) ? 2 : 3; ma = ay; sc = x;                 tc = (y > 0.f) ? z : -z; }
  else           { face = (z > 0.f) ? 4 : 5; ma = az; sc = (z > 0.f) ? x : -x; tc = -y; }
  float inv = 1.0f / fmaxf(ma, 1e-12f);
  u = 0.5f * (sc * inv + 1.0f) * Rf - 0.5f;
  v = 0.5f * (tc * inv + 1.0f) * Rf - 0.5f;
}

// Bilinear cube sample. Works for LDS-resident (ds_load after addrspace
// inference) and global (global_load) textures.
template <typename TP>
__device__ __forceinline__ F3 sample_cube(TP tex, int R,
                                          float dx, float dy, float dz) {
  int face; float u, v;
  cube_uv(dx, dy, dz, (float)R, face, u, v);
  float fx0 = floorf(u), fy0 = floorf(v);
  float wx = u - fx0, wy = v - fy0;
  int x0 = (int)fx0, y0 = (int)fy0;
  int x0i = min(max(x0,     0), R - 1);
  int x1i = min(max(x0 + 1, 0), R - 1);
  int y0i = min(max(y0,     0), R - 1);
  int y1i = min(max(y0 + 1, 0), R - 1);
  int base = face * R * R;
  int i00 = (base + y0i * R + x0i) * 3;
  int i01 = (base + y0i * R + x1i) * 3;
  int i10 = (base + y1i * R + x0i) * 3;
  int i11 = (base + y1i * R + x1i) * 3;
  float w00 = (1.f - wy) * (1.f - wx);
  float w01 = (1.f - wy) * wx;
  float w10 = wy * (1.f - wx);
  float w11 = wy * wx;
  F3 o;
  o.x = w00 * tex[i00+0] + w01 * tex[i01+0] + w10 * tex[i10+0] + w11 * tex[i11+0];
  o.y = w00 * tex[i00+1] + w01 * tex[i01+1] + w10 * tex[i10+1] + w11 * tex[i11+1];
  o.z = w00 * tex[i00+2] + w01 * tex[i01+2] + w10 * tex[i10+2] + w11 * tex[i11+2];
  return o;
}

// Branchless mip-pointer select (v_cndmask chain, no scratch, no divergence).
__device__ __forceinline__ const float* pick_mip(int l,
    const float* s0, const float* s1, const float* s2,
    const float* s3, const float* s4, const float* s5) {
  const float* t = s5;
  t = (l == 0) ? s0 : t;
  t = (l == 1) ? s1 : t;
  t = (l == 2) ? s2 : t;
  t = (l == 3) ? s3 : t;
  t = (l == 4) ? s4 : t;
  return t;
}

__global__ __launch_bounds__(256)
void EnvLight_kernel(const float* __restrict__ normal,
                     const float* __restrict__ refl,
                     const float* __restrict__ rough,
                     const float* __restrict__ diffuse,
                     const float* __restrict__ s0, const float* __restrict__ s1,
                     const float* __restrict__ s2, const float* __restrict__ s3,
                     const float* __restrict__ s4, const float* __restrict__ s5,
                     float* __restrict__ out_diff,
                     float* __restrict__ out_spec,
                     int N) {
  // ---- Stage the 18 KB diffuse cubemap into LDS with ONE Tensor Data Mover
  // DMA (TENSORcnt tracked): D# describes a 1-D tile of 4608 dword elements.
  __shared__ __align__(16) float sdiff[DIFF_FLOATS];
  if (threadIdx.x < 32) {   // wave 0 issues the DMA; EXEC ignored by TDM op
    uint32_t lds0 = (uint32_t)(uintptr_t)(void*)&sdiff[0]; // flat->LDS offset (addr[31:0])
    lds0 = __builtin_amdgcn_readfirstlane(lds0);
    uint64_t ga = (uint64_t)(uintptr_t)diffuse;
    const uint32_t ndw = DIFF_FLOATS;                      // elements (data_size = 4B)

    // D# group 0 (4 SGPRs): count=1 | lds_addr | global_addr[56:0] | type=2
    uint32_t g0w0 = 1u;
    uint32_t g0w1 = lds0;
    uint32_t g0w2 = (uint32_t)ga;
    uint32_t g0w3 = ((uint32_t)(ga >> 32) & 0x01FFFFFFu) | (2u << 30);

    // D# group 1 (8 SGPRs): mask=0, data_size=4B, no pad/iterate/barrier,
    // tensor_dim0 = tile_dim0 = ndw, tensor_dim1 = tile_dim1 = 1,
    // tensor_dim0_stride = ndw, tensor_dim1_stride = 0.
    uint32_t g1w0 = (2u << 16);                            // data_size=2 -> 4 bytes
    uint32_t g1w1 = (ndw & 0xFFFFu) << 16;                 // tensor_dim0[15:0]  @bits63:48
    uint32_t g1w2 = ((ndw >> 16) & 0xFFFFu) | (1u << 16);  // tensor_dim0[31:16] | tensor_dim1=1
    uint32_t g1w3 = (ndw & 0xFFFFu) << 16;                 // tile_dim0 @bits127:112
    uint32_t g1w4 = 1u;                                    // tile_dim1=1, tile_dim2=0
    uint32_t g1w5 = ndw;                                   // tensor_dim0_stride[31:0]
    uint32_t g1w6 = 0u;                                    // stride hi / dim1_stride lo
    uint32_t g1w7 = 0u;                                    // dim1_stride hi

    asm volatile("tensor_load_to_lds s[64:67], s[72:79]"
                 :
                 : "{s64}"(g0w0), "{s65}"(g0w1), "{s66}"(g0w2), "{s67}"(g0w3),
                   "{s72}"(g1w0), "{s73}"(g1w1), "{s74}"(g1w2), "{s75}"(g1w3),
                   "{s76}"(g1w4), "{s77}"(g1w5), "{s78}"(g1w6), "{s79}"(g1w7)
                 : "memory");
    __builtin_amdgcn_s_wait_tensorcnt(0);                  // DMA complete
  }
  __syncthreads();                                         // LDS tile visible to all waves

  int i = blockIdx.x * blockDim.x + threadIdx.x;
  if (i >= N) return;

  // ---- diffuse IBL from LDS-resident cubemap
  float nx = normal[i*3+0], ny = normal[i*3+1], nz = normal[i*3+2];
  F3 dl = sample_cube((const float*)sdiff, DIFF_RES, nx, ny, nz);
  out_diff[i*3+0] = dl.x;
  out_diff[i*3+1] = dl.y;
  out_diff[i*3+2] = dl.z;

  // ---- mip level from roughness (piecewise, matches get_mip)
  float r = rough[i];
  float level;
  if (r < MAXR)
    level = (fminf(fmaxf(r, MINR), MAXR) - MINR) * ((float)(NLEV - 2) / (MAXR - MINR));
  else
    level = (fminf(fmaxf(r, MAXR), 1.0f) - MAXR) * (1.0f / (1.0f - MAXR)) + (float)(NLEV - 2);
  level = fminf(fmaxf(level, 0.0f), (float)(NLEV - 1));
  int l0 = (int)floorf(level);
  l0 = min(max(l0, 0), NLEV - 1);
  int l1 = min(l0 + 1, NLEV - 1);
  float f = level - (float)l0;

  // ---- specular IBL: two adjacent-mip bilinear samples (== reference einsum)
  float rx = refl[i*3+0], ry = refl[i*3+1], rz = refl[i*3+2];
  const float* tA = pick_mip(l0, s0, s1, s2, s3, s4, s5);
  const float* tB = pick_mip(l1, s0, s1, s2, s3, s4, s5);
  int RA = 512 >> l0;
  int RB = 512 >> l1;
  F3 sA = sample_cube(tA, RA, rx, ry, rz);
  F3 sB = sample_cube(tB, RB, rx, ry, rz);
  float wA = 1.0f - f, wB = f;
  out_spec[i*3+0] = wA * sA.x + wB * sB.x;
  out_spec[i*3+1] = wA * sA.y + wB * sB.y;
  out_spec[i*3+2] = wA * sA.z + wB * sB.z;
}

extern "C" void kernel_launch(void* const* d_in, const int* in_sizes, int n_in,
                              void* d_out, int out_size, void* d_ws, size_t ws_size,
                              hipStream_t stream) {
  (void)n_in; (void)d_ws; (void)ws_size; (void)out_size;
  const float* normal  = (const float*)d_in[0];  // [B,H,W,3]
  const float* refl    = (const float*)d_in[1];  // [B,H,W,3]
  const float* rough   = (const float*)d_in[2];  // [B,H,W,1]
  const float* diffuse = (const float*)d_in[3];  // [6,16,16,3]
  const float* s0 = (const float*)d_in[4];       // [6,512,512,3]
  const float* s1 = (const float*)d_in[5];
  const float* s2 = (const float*)d_in[6];
  const float* s3 = (const float*)d_in[7];
  const float* s4 = (const float*)d_in[8];
  const float* s5 = (const float*)d_in[9];       // [6,16,16,3]

  int N = in_sizes[0] / 3;                       // B*H*W pixels
  float* out_diff = (float*)d_out;               // first output, flat
  float* out_spec = (float*)d_out + (size_t)N * 3; // second output, flat

  int block = 256;
  int grid = (N + block - 1) / block;
  EnvLight_kernel<<<grid, block, 0, stream>>>(normal, refl, rough, diffuse,
                                              s0, s1, s2, s3, s4, s5,
                                              out_diff, out_spec, N);
}